// MeanAggregator_36155034698032
// MI455X (gfx1250) — compile-verified
//
#include <hip/hip_runtime.h>

#define N_NODES 50000
#define N_EDGES 1600000
#define DIM     128
#define EPS     1e-8f

typedef __attribute__((ext_vector_type(2))) float v2f;
typedef __attribute__((ext_vector_type(8))) float v8f;

// ---------------------------------------------------------------------------
// GEMM: seq_fts[n][o] = sum_k seq[n][k] * W[o][k]    (seq @ W^T)
// Block = 256 threads = 8 wave32s; block handles 16 node-rows x all 128 cols.
// Wave w computes the 16x16 output tile at column w*16 with V_WMMA_F32_16X16X4_F32.
// ---------------------------------------------------------------------------
#define LDSW 132  // padded row stride (floats) -> bank-conflict-free A reads

__global__ void gemm_wmma_kernel(const float* __restrict__ seq,
                                 const float* __restrict__ W,
                                 float* __restrict__ seq_fts) {
    __shared__ float smem[16 * LDSW];  // 16 x 128 seq tile, padded

    const int tid     = threadIdx.x;
    const int rowBase = blockIdx.x * 16;

    // Stage 16x128 seq tile into LDS, coalesced float4 loads (512 float4s / 256 thr)
    for (int v = tid; v < 512; v += 256) {
        const int r  = v >> 5;        // 0..15
        const int c4 = v & 31;        // 0..31 (float4 index)
        float4 d = ((const float4*)(seq + (size_t)(rowBase + r) * DIM))[c4];
        *(float4*)&smem[r * LDSW + c4 * 4] = d;
    }
    __syncthreads();

    const int wave    = tid >> 5;      // 0..7 -> output column tile
    const int lane    = tid & 31;
    const int lo      = lane & 15;     // M (A) / N (B,C,D)
    const int hi      = lane >> 4;     // half-wave selector
    const int colBase = wave * 16;

    // A fragment source: lane lo = row M, K = k0 + 2*hi + {0,1}
    const float* arow = &smem[lo * LDSW + 2 * hi];
    // B fragment: B[k][n] = W[n][k]; lane lo = col N, K = k0 + 2*hi + {0,1}
    const float* brow = W + (size_t)(colBase + lo) * DIM + 2 * hi;

    v8f acc = {};
#pragma unroll
    for (int k0 = 0; k0 < DIM; k0 += 4) {
        v2f a = *(const v2f*)(arow + k0);  // 8B aligned (even float offset)
        v2f b = *(const v2f*)(brow + k0);
        acc = __builtin_amdgcn_wmma_f32_16x16x4_f32(
            /*neg_a=*/false, a, /*neg_b=*/false, b,
            /*c_mod=*/(short)0, acc, /*reuse_a=*/false, /*reuse_b=*/false);
    }

    // D layout: VGPR j holds M = j + 8*hi, N = lo
#pragma unroll
    for (int j = 0; j < 8; ++j) {
        const int m = j + 8 * hi;
        seq_fts[(size_t)(rowBase + m) * DIM + colBase + lo] = acc[j];
    }
}

// ---------------------------------------------------------------------------
// CSR construction: histogram -> 2-level exclusive scan -> scatter fill
// ---------------------------------------------------------------------------
#define SCAN_B  256
#define SCAN_NB ((N_NODES + SCAN_B - 1) / SCAN_B)  // 196

__global__ void zero_kernel(int* counts) {
    int i = blockIdx.x * blockDim.x + threadIdx.x;
    if (i < N_NODES) counts[i] = 0;
}

__global__ void hist_kernel(const int* __restrict__ rows, int* __restrict__ counts) {
    int e = blockIdx.x * blockDim.x + threadIdx.x;
    if (e < N_EDGES) atomicAdd(&counts[rows[e]], 1);
}

__global__ void partials_kernel(const int* __restrict__ counts, int* __restrict__ partials) {
    __shared__ int s[SCAN_B];
    int t = threadIdx.x;
    int i = blockIdx.x * SCAN_B + t;
    s[t] = (i < N_NODES) ? counts[i] : 0;
    __syncthreads();
    for (int off = SCAN_B / 2; off > 0; off >>= 1) {
        if (t < off) s[t] += s[t + off];
        __syncthreads();
    }
    if (t == 0) partials[blockIdx.x] = s[0];
}

__global__ void scan_top_kernel(const int* __restrict__ partials, int* __restrict__ blockoff) {
    __shared__ int s[SCAN_B];
    int t = threadIdx.x;
    s[t] = (t < SCAN_NB) ? partials[t] : 0;
    __syncthreads();
    for (int off = 1; off < SCAN_B; off <<= 1) {
        int x = (t >= off) ? s[t - off] : 0;
        __syncthreads();
        s[t] += x;
        __syncthreads();
    }
    if (t < SCAN_NB) blockoff[t] = (t == 0) ? 0 : s[t - 1];
}

__global__ void scan_write_kernel(const int* __restrict__ counts,
                                  const int* __restrict__ blockoff,
                                  int* __restrict__ rowptr,
                                  int* __restrict__ cursor) {
    __shared__ int s[SCAN_B];
    int t = threadIdx.x;
    int i = blockIdx.x * SCAN_B + t;
    int v = (i < N_NODES) ? counts[i] : 0;
    s[t] = v;
    __syncthreads();
    for (int off = 1; off < SCAN_B; off <<= 1) {
        int x = (t >= off) ? s[t - off] : 0;
        __syncthreads();
        s[t] += x;
        __syncthreads();
    }
    if (i < N_NODES) {
        int excl = s[t] - v + blockoff[blockIdx.x];
        rowptr[i] = excl;
        cursor[i] = excl;
    }
}

__global__ void fill_kernel(const int* __restrict__ rows, const int* __restrict__ cols,
                            int* __restrict__ cursor, int* __restrict__ csr_cols) {
    int e = blockIdx.x * blockDim.x + threadIdx.x;
    if (e < N_EDGES) {
        int pos = atomicAdd(&cursor[rows[e]], 1);
        csr_cols[pos] = cols[e];
    }
}

// ---------------------------------------------------------------------------
// Aggregation + mean + PReLU. One wave32 per node; lane holds float4 of
// features -> each edge is one coalesced 512B row load (L2 resident).
// No float atomics anywhere.
// ---------------------------------------------------------------------------
__global__ void agg_kernel(const float* __restrict__ seq_fts,
                           const int* __restrict__ rowptr,
                           const int* __restrict__ counts,
                           const int* __restrict__ csr_cols,
                           const float* __restrict__ alpha,
                           float* __restrict__ out) {
    const int wave = threadIdx.x >> 5;           // 0..3
    const int lane = threadIdx.x & 31;
    const int node = blockIdx.x * 4 + wave;      // 12500 * 4 == 50000 exact

    const int start = rowptr[node];
    const int len   = counts[node];

    const float* base = seq_fts + lane * 4;
    float4 acc = make_float4(0.f, 0.f, 0.f, 0.f);
    for (int j = 0; j < len; ++j) {
        const int c = csr_cols[start + j];       // broadcast load per wave
        float4 v = *(const float4*)(base + (size_t)c * DIM);
        acc.x += v.x; acc.y += v.y; acc.z += v.z; acc.w += v.w;
    }

    const float d = (float)len + EPS;
    const float a = alpha[0];
    float4 m;
    m.x = acc.x / d; m.y = acc.y / d; m.z = acc.z / d; m.w = acc.w / d;
    m.x = (m.x >= 0.f) ? m.x : a * m.x;
    m.y = (m.y >= 0.f) ? m.y : a * m.y;
    m.z = (m.z >= 0.f) ? m.z : a * m.z;
    m.w = (m.w >= 0.f) ? m.w : a * m.w;
    *(float4*)(out + (size_t)node * DIM + lane * 4) = m;
}

// ---------------------------------------------------------------------------
extern "C" void kernel_launch(void* const* d_in, const int* in_sizes, int n_in,
                              void* d_out, int out_size, void* d_ws, size_t ws_size,
                              hipStream_t stream) {
    (void)in_sizes; (void)n_in; (void)out_size; (void)ws_size;

    const float* seq   = (const float*)d_in[0];
    const float* W     = (const float*)d_in[1];
    const float* alpha = (const float*)d_in[2];
    const int*   rows  = (const int*)d_in[3];
    const int*   cols  = (const int*)d_in[4];
    float* out = (float*)d_out;

    // Workspace layout (all 16B aligned), total ~32.6 MB
    char* ws = (char*)d_ws;
    float* seq_fts  = (float*)(ws);                    // 25,600,000 B
    int*   counts   = (int*)(ws + 25600000);           //    200,000 B
    int*   rowptr   = (int*)(ws + 25800000);           //    200,000 B
    int*   cursor   = (int*)(ws + 26000000);           //    200,000 B
    int*   partials = (int*)(ws + 26200000);           //      1,024 B
    int*   blockoff = (int*)(ws + 26201024);           //      1,024 B
    int*   csr_cols = (int*)(ws + 26202048);           //  6,400,000 B

    const int edgeBlocks = (N_EDGES + 255) / 256;
    const int nodeBlocks = (N_NODES + 255) / 256;

    // 1) GEMM with WMMA (independent of CSR build)
    gemm_wmma_kernel<<<N_NODES / 16, 256, 0, stream>>>(seq, W, seq_fts);

    // 2) CSR build
    zero_kernel<<<nodeBlocks, 256, 0, stream>>>(counts);
    hist_kernel<<<edgeBlocks, 256, 0, stream>>>(rows, counts);
    partials_kernel<<<SCAN_NB, SCAN_B, 0, stream>>>(counts, partials);
    scan_top_kernel<<<1, SCAN_B, 0, stream>>>(partials, blockoff);
    scan_write_kernel<<<SCAN_NB, SCAN_B, 0, stream>>>(counts, blockoff, rowptr, cursor);
    fill_kernel<<<edgeBlocks, 256, 0, stream>>>(rows, cols, cursor, csr_cols);

    // 3) Gather-sum + mean + PReLU (one wave32 per node)
    agg_kernel<<<N_NODES / 4, 128, 0, stream>>>(seq_fts, rowptr, counts, csr_cols,
                                                alpha, out);
}